// FixedPointSolver_62474594288240
// MI455X (gfx1250) — compile-verified
//
#include <hip/hip_runtime.h>
#include <math.h>

#define GB 128      // graphs per batch
#define NN 512      // nodes per graph
#define NITER 64    // fixed launch count; real convergence ~20 iters (flag short-circuits rest)
#define TOLF 1e-6f

typedef float v2f __attribute__((ext_vector_type(2)));
typedef float v8f __attribute__((ext_vector_type(8)));

static const size_t M_FLOATS = (size_t)GB * NN * NN;  // 33,554,432 floats = 128 MiB
static const size_t Z_FLOATS = (size_t)GB * NN;       // 65,536

// Swizzled index so that M is stored exactly in the V_WMMA_F32_16X16X4_F32
// A-operand lane layout. Tiles are 16 rows x 4 K; 4 tiles grouped (K=16) so each
// lane's data for a group is 8 contiguous floats (two b128 loads).
// A layout (ISA 7.12.2): lane = m + 16*(k>>1); VGPR slot = k&1.
__device__ __forceinline__ size_t swz_index(int g, int row, int col) {
  int rb   = row >> 4;          // 16-row block
  int m    = row & 15;
  int kc4  = col >> 4;          // K=16 group
  int t    = (col >> 2) & 3;    // tile within group
  int k    = col & 3;           // k within tile
  int lane = m + ((k & 2) << 3);
  int slot = k & 1;
  return ((size_t)((g * 32 + rb) * 32 + kc4)) * 256 + (size_t)(lane * 8 + t * 2 + slot);
}

__global__ void fp_init(float* __restrict__ M, float* __restrict__ z0,
                        unsigned* __restrict__ diffs, unsigned* __restrict__ flag) {
  size_t i = (size_t)blockIdx.x * blockDim.x + threadIdx.x;
  size_t stride = (size_t)gridDim.x * blockDim.x;
  float4 zero = make_float4(0.f, 0.f, 0.f, 0.f);
  size_t n4 = M_FLOATS >> 2;
  for (size_t j = i; j < n4; j += stride) ((float4*)M)[j] = zero;
  if (i < Z_FLOATS) z0[i] = ((i & (NN - 1)) == 0) ? 1.0f : 0.0f;  // b = sink mask
  if (i < NITER) diffs[i] = 0u;
  if (i == 0) *flag = 0u;
}

__global__ void fp_build(const float* __restrict__ utils, const int* __restrict__ ei,
                         int E, float* __restrict__ M) {
  int e = blockIdx.x * blockDim.x + threadIdx.x;
  if (e >= E) return;
  int s = ei[e];          // edge_index[0][e]
  int d = ei[E + e];      // edge_index[1][e]  (same graph as s)
  int g   = s >> 9;       // N = 512
  int row = s & (NN - 1);
  int col = d & (NN - 1);
  atomicAdd(&M[swz_index(g, row, col)], expf(utils[e]));
}

// One wave per (graph, 16-row block). z staged in LDS per workgroup.
// y(16) = sum over 128 WMMA_F32_16x16x4_F32 tiles; z replicated into all 16 B cols.
__global__ __launch_bounds__(256) void fp_step(const float* __restrict__ M,
                                               const float* __restrict__ zin,
                                               float* __restrict__ zout,
                                               unsigned* __restrict__ diff_slot,
                                               const unsigned* __restrict__ flag) {
  __shared__ __align__(16) float zlds[NN];
  __shared__ float wmax[8];
  const int g   = blockIdx.x >> 2;   // 4 workgroups per graph
  const int rbg = blockIdx.x & 3;    // 8 row-blocks each
  const int tid = threadIdx.x;

  if (*flag) {  // converged at an earlier launch: carry z_K forward unchanged
    int base = g * NN + rbg * 128;
    if (tid < 128) zout[base + tid] = zin[base + tid];
    return;
  }

  zlds[tid]        = zin[g * NN + tid];
  zlds[tid + 256]  = zin[g * NN + 256 + tid];
  __syncthreads();

  const int wid  = tid >> 5;
  const int lane = tid & 31;
  const int rb   = rbg * 8 + wid;
  const float* mp = M + ((size_t)((g * 32 + rb) * 32)) * 256 + (size_t)(lane * 8);
  const int h2 = (lane >> 4) << 1;   // B operand: lanes 0-15 hold K0,K1; 16-31 hold K2,K3

  v8f acc = {0.f, 0.f, 0.f, 0.f, 0.f, 0.f, 0.f, 0.f};
  #pragma unroll 4
  for (int kc4 = 0; kc4 < 32; ++kc4) {
    const float4 a0 = *(const float4*)(mp + (size_t)kc4 * 256);
    const float4 a1 = *(const float4*)(mp + (size_t)kc4 * 256 + 4);
    const int cb = kc4 * 16;
    const float2 b0 = *(const float2*)&zlds[cb + 0  + h2];
    const float2 b1 = *(const float2*)&zlds[cb + 4  + h2];
    const float2 b2 = *(const float2*)&zlds[cb + 8  + h2];
    const float2 b3 = *(const float2*)&zlds[cb + 12 + h2];
    v2f a, b;
    a.x = a0.x; a.y = a0.y; b.x = b0.x; b.y = b0.y;
    acc = __builtin_amdgcn_wmma_f32_16x16x4_f32(false, a, false, b, (short)0, acc, false, false);
    a.x = a0.z; a.y = a0.w; b.x = b1.x; b.y = b1.y;
    acc = __builtin_amdgcn_wmma_f32_16x16x4_f32(false, a, false, b, (short)0, acc, false, false);
    a.x = a1.x; a.y = a1.y; b.x = b2.x; b.y = b2.y;
    acc = __builtin_amdgcn_wmma_f32_16x16x4_f32(false, a, false, b, (short)0, acc, false, false);
    a.x = a1.z; a.y = a1.w; b.x = b3.x; b.y = b3.y;
    acc = __builtin_amdgcn_wmma_f32_16x16x4_f32(false, a, false, b, (short)0, acc, false, false);
  }

  // D layout: lane l, VGPR v -> row (v + 8*(l>>4)), all 16 columns identical.
  const int j = lane & 7;
  float y = acc[0];
  if (j == 1) y = acc[1];
  if (j == 2) y = acc[2];
  if (j == 3) y = acc[3];
  if (j == 4) y = acc[4];
  if (j == 5) y = acc[5];
  if (j == 6) y = acc[6];
  if (j == 7) y = acc[7];
  const int rloc = j + ((lane >> 4) << 3);
  const int lrow = rb * 16 + rloc;
  const float zn = y + ((lrow == 0) ? 1.0f : 0.0f);  // + b (sink = local node 0)

  float d = 0.0f;
  if ((lane & 15) < 8) {   // lanes 0-7 write rows 0-7, lanes 16-23 write rows 8-15
    d = fabsf(zn - zlds[lrow]);
    zout[g * NN + lrow] = zn;
  }
  #pragma unroll
  for (int off = 16; off > 0; off >>= 1)
    d = fmaxf(d, __shfl_xor(d, off, 32));
  if (lane == 0) wmax[wid] = d;
  __syncthreads();
  if (tid == 0) {
    float m = wmax[0];
    #pragma unroll
    for (int w = 1; w < 8; ++w) m = fmaxf(m, wmax[w]);
    atomicMax(diff_slot, __float_as_uint(m));  // nonneg float bits are order-preserving
  }
}

__global__ void fp_check(const unsigned* __restrict__ diff, unsigned* __restrict__ flag) {
  if (__uint_as_float(*diff) <= TOLF) *flag = 1u;
}

__global__ void fp_final(const float* __restrict__ z, float* __restrict__ out, int n) {
  int i = blockIdx.x * blockDim.x + threadIdx.x;
  if (i < n) {
    float zi = z[i];
    out[i] = (zi > 0.0f) ? logf(zi) : __builtin_nanf("");  // log(0) = -inf -> NaN
  }
}

extern "C" void kernel_launch(void* const* d_in, const int* in_sizes, int n_in,
                              void* d_out, int out_size, void* d_ws, size_t ws_size,
                              hipStream_t stream) {
  const float* utils = (const float*)d_in[0];
  const int*   ei    = (const int*)d_in[1];   // edge_index flattened (2, E), int32
  const int    E     = in_sizes[0];

  char* ws = (char*)d_ws;
  float*    M     = (float*)ws;
  float*    zb0   = (float*)(ws + M_FLOATS * sizeof(float));
  float*    zb1   = zb0 + Z_FLOATS;
  unsigned* diffs = (unsigned*)(zb1 + Z_FLOATS);
  unsigned* flag  = diffs + NITER;

  fp_init<<<4096, 256, 0, stream>>>(M, zb0, diffs, flag);
  fp_build<<<(E + 255) / 256, 256, 0, stream>>>(utils, ei, E, M);

  float* bufs[2] = {zb0, zb1};
  for (int it = 1; it <= NITER; ++it) {
    const float* zi = bufs[(it + 1) & 1];
    float*       zo = bufs[it & 1];
    fp_step<<<GB * 4, 256, 0, stream>>>(M, zi, zo, diffs + (it - 1), flag);
    fp_check<<<1, 1, 0, stream>>>(diffs + (it - 1), flag);
  }
  fp_final<<<((int)Z_FLOATS + 255) / 256, 256, 0, stream>>>(bufs[NITER & 1], (float*)d_out,
                                                            (int)Z_FLOATS);
}